// LambdaRankLoss_34084860461249
// MI455X (gfx1250) — compile-verified
//
#include <hip/hip_runtime.h>
#include <hip/hip_bf16.h>
#include <math.h>

// ---------------------------------------------------------------------------
// LambdaRank loss for MI455X (gfx1250, wave32).
//
// Sparsified exact algorithm: pairs with both docs outside the top-16 by
// score have disc_i == disc_j == 0 => zero contribution. Per group we
// evaluate only (top16 x all) + (all x top16) - (top16 x top16) = 33 tiles
// of 16x16 instead of 256 tiles.
//
// Each tile's three difference matrices (pdiff / ydiff / ddiff) are rank-2
// outer products computed with V_WMMA_F32_16X16X4_F32:
//     D = A x B,  A(16x4) = [row_vals, ones, 0, 0],  B(4x16) = [ones; -col_vals; 0; 0]
// landing all three in the same C/D accumulator layout so the nonlinear
// weighting + logistic loss is purely elementwise on accumulator VGPRs.
//
// Ranking uses packed 64-bit keys (sortable(p) << 32 | (255 - idx)) so each
// pairwise comparison is one v_cmp_gt_u64, matching jnp.argsort(-p) stable
// tie-break semantics exactly.
// ---------------------------------------------------------------------------

#define WAVES_PER_BLOCK 8
#define GROUP_L 256
#define K_NDCG 10
#define SIGMA_F 1.0f
#define EPS_F 1e-10f

typedef __attribute__((ext_vector_type(2))) float v2f;
typedef __attribute__((ext_vector_type(8))) float v8f;

#if defined(__gfx1250__) && __has_builtin(__builtin_amdgcn_wmma_f32_16x16x4_f32)
#define USE_WMMA 1
#else
#define USE_WMMA 0
#endif

__device__ __forceinline__ float fast_rcp(float x) {
#if defined(__gfx1250__) && __has_builtin(__builtin_amdgcn_rcpf)
  return __builtin_amdgcn_rcpf(x);  // v_rcp_f32, ~1 ULP
#else
  return 1.0f / x;
#endif
}

// Order-preserving float -> uint transform (ascending).
__device__ __forceinline__ unsigned int f32_sortable(float f) {
  unsigned int u = __float_as_uint(f);
  return u ^ ((u & 0x80000000u) ? 0xFFFFFFFFu : 0x80000000u);
}

// Sum over a 16x16 tile of  w(i,j) * pairwise logistic loss.
// rP/rY/rD: 16 row values (LDS), cP/cY/cD: 16 col values (LDS).
__device__ __forceinline__ float tile_loss(const float* rP, const float* rY,
                                           const float* rD, const float* cP,
                                           const float* cY, const float* cD,
                                           float inv_idcg, int lane) {
  const int s = lane & 15;
  float f_pd[8], f_yd[8], f_dd[8];
#if USE_WMMA
  {
    const bool lo = lane < 16;  // lanes 0-15 hold K=0,1; lanes 16-31 K=2,3 (zeroed)
    const float one = lo ? 1.0f : 0.0f;
    v2f a_p = {lo ? rP[s] : 0.0f, one};
    v2f a_y = {lo ? rY[s] : 0.0f, one};
    v2f a_d = {lo ? rD[s] : 0.0f, one};
    v2f b_p = {one, lo ? -cP[s] : 0.0f};
    v2f b_y = {one, lo ? -cY[s] : 0.0f};
    v2f b_d = {one, lo ? -cD[s] : 0.0f};
    v8f z = {0.0f, 0.0f, 0.0f, 0.0f, 0.0f, 0.0f, 0.0f, 0.0f};
    // D[M][N] = row[M]*1 + 1*(-col[N]) = row[M] - col[N]
    v8f PD = __builtin_amdgcn_wmma_f32_16x16x4_f32(false, a_p, false, b_p,
                                                   (short)0, z, false, false);
    v8f YD = __builtin_amdgcn_wmma_f32_16x16x4_f32(false, a_y, false, b_y,
                                                   (short)0, z, false, false);
    v8f DD = __builtin_amdgcn_wmma_f32_16x16x4_f32(false, a_d, false, b_d,
                                                   (short)0, z, false, false);
#pragma unroll
    for (int k = 0; k < 8; ++k) {
      f_pd[k] = PD[k];
      f_yd[k] = YD[k];
      f_dd[k] = DD[k];
    }
  }
#else
  {
    // VALU fallback: lane covers col s, rows k + 8*(lane>>4).
    const int half = lane >> 4;
#pragma unroll
    for (int k = 0; k < 8; ++k) {
      const int r = k + 8 * half;
      f_pd[k] = rP[r] - cP[s];
      f_yd[k] = rY[r] - cY[s];
      f_dd[k] = rD[r] - cD[s];
    }
  }
#endif
  float acc = 0.0f;
#pragma unroll
  for (int k = 0; k < 8; ++k) {
    // gate y_i > y_j: max(yd,0) == yd under the gate, 0 otherwise
    const float wgt = fmaxf(f_yd[k], 0.0f) * fabsf(f_dd[k]) * inv_idcg;
    // sigmoid via v_exp + v_rcp (no IEEE division chain)
    const float sg = fast_rcp(1.0f + __expf(-SIGMA_F * f_pd[k]));
    acc = fmaf(wgt, -__logf(sg + EPS_F), acc);
  }
  return acc;
}

__global__ __launch_bounds__(32 * WAVES_PER_BLOCK) void lambdarank_kernel(
    const float* __restrict__ pred, const int* __restrict__ lab,
    float* __restrict__ out, int G, float invG) {
  __shared__ float sP[WAVES_PER_BLOCK][GROUP_L];
  __shared__ float sY[WAVES_PER_BLOCK][GROUP_L];
  __shared__ float sD[WAVES_PER_BLOCK][GROUP_L];
  __shared__ unsigned long long sK[WAVES_PER_BLOCK][GROUP_L];
  __shared__ float sPT[WAVES_PER_BLOCK][16];
  __shared__ float sYT[WAVES_PER_BLOCK][16];
  __shared__ float sDT[WAVES_PER_BLOCK][16];

  const int lane = threadIdx.x & 31;
  const int w = threadIdx.x >> 5;
  const int g = blockIdx.x * WAVES_PER_BLOCK + w;
  const bool active = (g < G);

  float pi[8];
  float yi[8];
  unsigned long long key[8];
  int cnt1 = 0, cnt2 = 0, cnt3 = 0, cnt4 = 0;

  // ---- Phase 1: load scores/labels, build sort keys, label histogram ----
  if (active) {
    const float* pg = pred + (size_t)g * GROUP_L;
    const int* lg = lab + (size_t)g * GROUP_L;
#pragma unroll
    for (int k = 0; k < 8; ++k) {
      const int i = lane + 32 * k;
      pi[k] = pg[i];
      const int lv = lg[i];
      yi[k] = (float)lv;
      cnt1 += (lv == 1);
      cnt2 += (lv == 2);
      cnt3 += (lv == 3);
      cnt4 += (lv == 4);
      // descending score, ascending index tie-break == jnp stable argsort(-p)
      key[k] = ((unsigned long long)f32_sortable(pi[k]) << 32) |
               (unsigned long long)(unsigned int)(GROUP_L - 1 - i);
      sP[w][i] = pi[k];
      sY[w][i] = yi[k];
      sK[w][i] = key[k];
    }
  }
  __syncthreads();

  // ---- Phase 2: ranks via 64-bit key compares, discounts, top-16 scatter --
  if (active) {
    int rank[8] = {0, 0, 0, 0, 0, 0, 0, 0};
    for (int j = 0; j < GROUP_L; ++j) {
      const unsigned long long kj = sK[w][j];
#pragma unroll
      for (int k = 0; k < 8; ++k) rank[k] += (int)(kj > key[k]);
    }
#pragma unroll
    for (int k = 0; k < 8; ++k) {
      const int i = lane + 32 * k;
      const float d =
          (rank[k] < K_NDCG) ? 1.0f / __log2f((float)rank[k] + 2.0f) : 0.0f;
      sD[w][i] = d;
      if (rank[k] < 16) {  // ranks are a permutation -> unique slots
        sPT[w][rank[k]] = pi[k];
        sYT[w][rank[k]] = yi[k];
        sDT[w][rank[k]] = d;
      }
    }
  }
  __syncthreads();

  float acc = 0.0f;
  if (active) {
    // ---- Phase 3: IDCG from label histogram (wave butterfly reduce) ----
#pragma unroll
    for (int off = 16; off >= 1; off >>= 1) {
      cnt1 += __shfl_xor(cnt1, off, 32);
      cnt2 += __shfl_xor(cnt2, off, 32);
      cnt3 += __shfl_xor(cnt3, off, 32);
      cnt4 += __shfl_xor(cnt4, off, 32);
    }
    float idcg = 0.0f;
    {
      int pos = 0;
      int cs[4] = {cnt4, cnt3, cnt2, cnt1};  // label values 4..1 descending
#pragma unroll
      for (int vi = 0; vi < 4; ++vi) {
        const float val = (float)(4 - vi);
        int c = cs[vi];
        if (c > K_NDCG - pos) c = K_NDCG - pos;
        for (int t = 0; t < c; ++t) {
          idcg += val / __log2f((float)pos + 2.0f);
          ++pos;
        }
      }
    }
    const float inv_idcg = (idcg > 0.0f) ? fast_rcp(idcg) : 0.0f;

    // ---- Phase 4: 33 tiles: (top16 x all) + (all x top16) - corner ----
    for (int jt = 0; jt < 16; ++jt) {
      acc += tile_loss(sPT[w], sYT[w], sDT[w], &sP[w][jt * 16], &sY[w][jt * 16],
                       &sD[w][jt * 16], inv_idcg, lane);
    }
    for (int it = 0; it < 16; ++it) {
      acc += tile_loss(&sP[w][it * 16], &sY[w][it * 16], &sD[w][it * 16],
                       sPT[w], sYT[w], sDT[w], inv_idcg, lane);
    }
    acc -= tile_loss(sPT[w], sYT[w], sDT[w], sPT[w], sYT[w], sDT[w], inv_idcg,
                     lane);

    // ---- Phase 5: wave reduction + one atomic per group ----
#pragma unroll
    for (int off = 16; off >= 1; off >>= 1) acc += __shfl_xor(acc, off, 32);
    if (lane == 0) atomicAdd(out, acc * invG);
  }
}

extern "C" void kernel_launch(void* const* d_in, const int* in_sizes, int n_in,
                              void* d_out, int out_size, void* d_ws,
                              size_t ws_size, hipStream_t stream) {
  const float* pred = (const float*)d_in[0];
  const int* lab = (const int*)d_in[1];
  float* out = (float*)d_out;

  const int N = in_sizes[0];
  const int G = N / GROUP_L;  // reference: 1024 groups of 256

  // Output is accumulated with atomics: zero it first (capture-legal memset).
  hipMemsetAsync(out, 0, sizeof(float) * (out_size > 0 ? out_size : 1),
                 stream);

  dim3 block(32 * WAVES_PER_BLOCK);
  dim3 grid((G + WAVES_PER_BLOCK - 1) / WAVES_PER_BLOCK);
  lambdarank_kernel<<<grid, block, 0, stream>>>(pred, lab, out, G,
                                                1.0f / (float)G);
}